// RNNTLossNumba_13657996002100
// MI455X (gfx1250) — compile-verified
//
#include <hip/hip_runtime.h>
#include <stdint.h>

// Problem constants (match the reference).
#define Bsz 8
#define Tn  256
#define Un  101
#define Vn  512
#define NEGF (-1e30f)

typedef uint32_t u32x4 __attribute__((ext_vector_type(4)));
typedef uint32_t u32x8 __attribute__((ext_vector_type(8)));

// ---------------------------------------------------------------------------
// CDNA5 async global->LDS copy (VGLOBAL encoding, tracked by ASYNCcnt).
// ---------------------------------------------------------------------------
__device__ __forceinline__ uint32_t lds_off_of(const void* p) {
  // shared -> generic addrspacecast yields {aperture_hi, lds_offset_lo32}
  return (uint32_t)(uintptr_t)p;
}

__device__ __forceinline__ void async_copy_b128(uint32_t lds_byte_addr, uint64_t gaddr) {
  asm volatile("global_load_async_to_lds_b128 %0, %1, off"
               :: "v"(lds_byte_addr), "v"(gaddr)
               : "memory");
}

__device__ __forceinline__ void wait_async_zero() {
  asm volatile("s_wait_asynccnt 0" ::: "memory");
}

// ---------------------------------------------------------------------------
// CDNA5 Tensor Data Mover: 1-D contiguous global->LDS DMA of nelem 4-byte
// elements. Builds the D# (descriptor) per CDNA5 ISA:
//   group0: count=1 | lds_addr | global_addr[56:0] | type=2
//   group1: data_size=4B, tensor_dim0=nelem, tensor_dim1=1,
//           tile_dim0=nelem (fits 16 bits), tile_dim1=1, dim0_stride=nelem
// Groups 2/3 = null (tensor rank <= 2). Tracked by TENSORcnt.
// ---------------------------------------------------------------------------
__device__ __forceinline__ void tdm_load_1d(uint64_t gaddr, uint32_t lds_byte,
                                            uint32_t nelem) {
  const u32x4 g0 = {
      0x1u,                                       // count=1, user mode
      lds_byte,                                   // lds_addr (bytes)
      (uint32_t)gaddr,                            // global_addr[31:0]
      (uint32_t)(gaddr >> 32) | (2u << 30)        // global_addr[56:32] | type=2
  };
  const u32x8 g1 = {
      2u << 16,                                   // data_size = 4 bytes
      (nelem & 0xFFFFu) << 16,                    // tensor_dim0[15:0]
      (nelem >> 16) | (1u << 16),                 // tensor_dim0[31:16], tensor_dim1=1
      (nelem & 0xFFFFu) << 16,                    // tile_dim0 = nelem (< 65536)
      1u,                                         // tile_dim1 = 1, tile_dim2 = 0
      nelem,                                      // tensor_dim0_stride[31:0]
      0u,                                         // stride0[47:32]=0, stride1[15:0]=0
      0u                                          // tensor_dim1_stride[47:16] = 0
  };
  asm volatile("tensor_load_to_lds %0, %1, null, null"
               :: "s"(g0), "s"(g1)
               : "memory");
}

__device__ __forceinline__ float logaddexpf_(float a, float b) {
  float m = fmaxf(a, b);
  float d = fabsf(a - b);
  return m + log1pf(__expf(-d));   // exp(-1e30)=0 -> returns m when one side is NEG
}

// ---------------------------------------------------------------------------
// Kernel 1: per-row (b,t,u) logsumexp over V=512, emit blankp / emit tables.
// One wave32 per row; 8 rows (= 8 waves = 256 threads) per block.
// Row is staged into LDS with async b128 copies; reductions via wave32 shfl.
// ---------------------------------------------------------------------------
__global__ __launch_bounds__(256) void rnnt_softmax_gather(
    const float* __restrict__ acts, const int* __restrict__ labels,
    float* __restrict__ bp, float* __restrict__ em) {
  __shared__ float rows[8 * Vn];                      // 16 KB
  const int wave = threadIdx.x >> 5;
  const int lane = threadIdx.x & 31;
  const int row  = blockIdx.x * 8 + wave;             // grid exactly covers B*T*U

  const uint64_t gbase = (uint64_t)(uintptr_t)acts + (uint64_t)row * (Vn * 4u);
  const uint32_t lbase = lds_off_of(&rows[wave * Vn]);

  // 512 floats = 128 x 16B chunks; 32 lanes x 4 async b128 each.
  #pragma unroll
  for (int i = 0; i < 4; ++i) {
    const uint32_t off = (uint32_t)(lane * 16 + i * 512);
    async_copy_b128(lbase + off, gbase + off);
  }
  wait_async_zero();   // own row only -> wave-local wait, no barrier needed

  // lane reads float4 chunks [lane + 32k], k=0..3 : contiguous 512B sweeps,
  // conflict-free across the 64 LDS banks.
  const float4* rv = (const float4*)(&rows[wave * Vn]);
  float4 v[4];
  float mx = -3.4e38f;
  #pragma unroll
  for (int k = 0; k < 4; ++k) {
    v[k] = rv[lane + 32 * k];
    mx = fmaxf(mx, fmaxf(fmaxf(v[k].x, v[k].y), fmaxf(v[k].z, v[k].w)));
  }
  #pragma unroll
  for (int off = 16; off >= 1; off >>= 1)
    mx = fmaxf(mx, __shfl_xor(mx, off, 32));

  float s = 0.f;
  #pragma unroll
  for (int k = 0; k < 4; ++k)
    s += __expf(v[k].x - mx) + __expf(v[k].y - mx) +
         __expf(v[k].z - mx) + __expf(v[k].w - mx);
  #pragma unroll
  for (int off = 16; off >= 1; off >>= 1)
    s += __shfl_xor(s, off, 32);

  const float lse = mx + __logf(s);

  if (lane == 0) {
    const uint32_t u  = (uint32_t)row % Un;
    const uint32_t bt = (uint32_t)row / Un;           // b*T + t
    bp[row] = rows[wave * Vn + 0] - lse;              // log P(blank | b,t,u)
    if (u < Un - 1) {
      const uint32_t b = bt / Tn;
      const int lab = labels[b * (Un - 1) + u];       // label in [1, V)
      em[bt * (Un - 1) + u] = rows[wave * Vn + lab] - lse;
    }
  }
}

// ---------------------------------------------------------------------------
// Kernel 2: per-batch alpha DP over the (T,U) lattice, anti-diagonal wavefront.
// Whole per-b blankp (T*U) + emit (T*(U-1)) slab (~201 KB) is DMA'd into LDS
// by the Tensor Data Mover (320 KB/WGP on CDNA5), then 356 diagonals run
// entirely from LDS, one workgroup barrier per diagonal.
// ---------------------------------------------------------------------------
__global__ __launch_bounds__(128) void rnnt_alpha_dp(
    const float* __restrict__ bp, const float* __restrict__ em,
    const int* __restrict__ act_lens, const int* __restrict__ label_lens,
    float* __restrict__ out) {
  extern __shared__ float slab[];                     // bp_l[T*U] ++ em_l[T*(U-1)]
  float* bp_l = slab;
  float* em_l = slab + Tn * Un;
  __shared__ float diag[2][Un + 3];
  __shared__ float resval;

  const int b   = blockIdx.x;
  const int tid = threadIdx.x;

  // ---- TDM fill of the per-b slabs: wave 0 issues two descriptor DMAs -----
  if (tid < 32) {                                     // whole-wave branch
    const uint64_t gbp = (uint64_t)(uintptr_t)bp + (uint64_t)b * (Tn * Un * 4u);
    const uint64_t gem = (uint64_t)(uintptr_t)em + (uint64_t)b * (Tn * (Un - 1) * 4u);
    tdm_load_1d(gbp, lds_off_of(bp_l), Tn * Un);          // 25856 elems
    tdm_load_1d(gem, lds_off_of(em_l), Tn * (Un - 1));    // 25600 elems
    __builtin_amdgcn_s_wait_tensorcnt(0);
  }
  __syncthreads();                                    // publish LDS to all waves

  const int tend = act_lens[b] - 1;
  const int uend = label_lens[b];
  const int dtar = tend + uend;

  // alpha[t,u] = logaddexp(alpha[t-1,u] + blank[t-1,u], alpha[t,u-1] + emit[t,u-1])
  // Cells on diagonal d = t+u depend only on diagonal d-1 -> ping-pong buffers.
  for (int d = 0; d < Tn + Un - 1; ++d) {
    const int lo = max(0, d - (Tn - 1));
    const int hi = min(d, Un - 1);
    const int u  = tid;
    if (u >= lo && u <= hi) {
      const int t = d - u;
      float val;
      if (d == 0) {
        val = 0.0f;                                   // alpha[0,0]
      } else {
        const float* prev = diag[(d - 1) & 1];
        const float a  = (t > 0) ? prev[u]     + bp_l[(t - 1) * Un + u]       : NEGF;
        const float bb = (u > 0) ? prev[u - 1] + em_l[t * (Un - 1) + (u - 1)] : NEGF;
        val = logaddexpf_(a, bb);
      }
      diag[d & 1][u] = val;
      if (d == dtar && u == uend) resval = val;       // alpha[tend, uend]
    }
    __syncthreads();
  }

  if (tid == 0)
    out[b] = -(resval + bp_l[tend * Un + uend]);
}

// ---------------------------------------------------------------------------
// Host-side launch.
//   d_in[0] acts       (B*T*U*V) float32
//   d_in[1] labels     (B*(U-1)) int32
//   d_in[2] act_lens   (B)       int32
//   d_in[3] label_lens (B)       int32
//   d_out   (B) float32
//   d_ws    >= (B*T*U + B*T*(U-1)) * 4 bytes (~1.65 MB)
// ---------------------------------------------------------------------------
extern "C" void kernel_launch(void* const* d_in, const int* in_sizes, int n_in,
                              void* d_out, int out_size, void* d_ws, size_t ws_size,
                              hipStream_t stream) {
  const float* acts       = (const float*)d_in[0];
  const int*   labels     = (const int*)d_in[1];
  const int*   act_lens   = (const int*)d_in[2];
  const int*   label_lens = (const int*)d_in[3];
  float*       out        = (float*)d_out;

  float* bp = (float*)d_ws;                       // B*T*U floats
  float* em = bp + (size_t)Bsz * Tn * Un;         // B*T*(U-1) floats

  const int nrows = Bsz * Tn * Un;                // 206848, divisible by 8
  rnnt_softmax_gather<<<nrows / 8, 256, 0, stream>>>(acts, labels, bp, em);

  const size_t dyn = (size_t)(Tn * Un + Tn * (Un - 1)) * sizeof(float); // ~201 KB
  (void)hipFuncSetAttribute((const void*)rnnt_alpha_dp,
                            hipFuncAttributeMaxDynamicSharedMemorySize, (int)dyn);
  rnnt_alpha_dp<<<Bsz, 128, dyn, stream>>>(bp, em, act_lens, label_lens, out);
}